// MeshLoss_2834678415375
// MI455X (gfx1250) — compile-verified
//
#include <hip/hip_runtime.h>

typedef __attribute__((ext_vector_type(2))) float v2f;
typedef __attribute__((ext_vector_type(8))) float v8f;

#define MPTS   16384          // query points per batch
#define NPTS   1024           // 32*32 top points per batch
#define NM_J   16
#define NM_K   32
#define FEM_COUNT (4*3*32*15*32)   // 184320
#define DIST_COUNT (4*16384)       // 65536

// -------- distance kernel: WMMA f32 16x16x4 ---------------------------------
// K=0..2 carry the 3-D dot product (A pre-scaled by -2); K=3 carries the
// ||t||^2 term (A=1, B=tn), so D = tn - 2*p.t with C = inline 0.
// 8 waves / block; block stages pre-swizzled B operands in LDS (16 KB):
//   s_pack[n]        (lanes 0-15, K=0,1) = {t0, t1}
//   s_pack[NPTS + n] (lanes 16-31, K=2,3) = {t2, tn}
__global__ __launch_bounds__(256) void MeshLoss_dist_kernel(
    const float* __restrict__ pc,       // (4,3,16384)
    const float* __restrict__ nm,       // (4,3,32,16,32)
    float* __restrict__ acc) {
  __shared__ v2f s_pack[2 * NPTS];      // 16 KB

  const int tid  = threadIdx.x;
  const int lane = tid & 31;
  const int wave = tid >> 5;

  const int blocksPerBatch = (MPTS / 16) / 8;           // 128
  const int b  = blockIdx.x / blocksPerBatch;
  const int mg = blockIdx.x % blocksPerBatch;

  // Stage top = nm[b, d, i, 15, k] with n = i*32 + k; pre-swizzle for WMMA B.
  for (int n = tid; n < NPTS; n += 256) {
    const int i = n >> 5, k = n & 31;
    const size_t base = ((size_t)(b * 3 + 0) * 32 + i) * (NM_J * NM_K) + 15 * NM_K + k;
    const float t0 = nm[base];
    const float t1 = nm[base + 16384];      // d-plane stride = 32*16*32
    const float t2 = nm[base + 2 * 16384];
    const float tn = t0 * t0 + t1 * t1 + t2 * t2;
    s_pack[n]        = (v2f){t0, t1};
    s_pack[NPTS + n] = (v2f){t2, tn};
  }
  __syncthreads();

  const int  m0   = (mg * 8 + wave) * 16;
  const int  mrow = lane & 15;
  const bool lo   = lane < 16;
  const int  m    = m0 + mrow;
  const float* pcb = pc + (size_t)b * 3 * MPTS;
  const float p0 = pcb[m];
  const float p1 = pcb[MPTS + m];
  const float p2 = pcb[2 * MPTS + m];
  const float pn_lane = p0 * p0 + p1 * p1 + p2 * p2;

  // A-matrix (16x4 f32): lanes 0-15 -> K=0,1 ; lanes 16-31 -> K=2,3.
  // K=3 column of A is all ones so it picks up B's tn row.
  v2f a;
  a.x = lo ? (-2.0f * p0) : (-2.0f * p2);
  a.y = lo ? (-2.0f * p1) : 1.0f;

  // Per-half B operand base pointer (computed once; loop is divergence-free).
  const v2f* __restrict__ bp = &s_pack[lo ? 0 : NPTS];

  float mmin[8];
#pragma unroll
  for (int v = 0; v < 8; ++v) mmin[v] = 3.0e38f;

#pragma unroll 4
  for (int nt = 0; nt < NPTS / 16; ++nt) {
    const v2f bm = bp[nt * 16 + mrow];      // single ds_load_b64
    v8f c = {};                             // C = inline 0
    c = __builtin_amdgcn_wmma_f32_16x16x4_f32(
        /*neg_a=*/false, a, /*neg_b=*/false, bm,
        /*c_mod=*/(short)0, c, /*reuse_a=*/false, /*reuse_b=*/false);
#pragma unroll
    for (int v = 0; v < 8; ++v)
      mmin[v] = fminf(mmin[v], c[v]);       // tn - 2 p.t
  }

  // Min-reduce each accumulator across its 16-lane half (columns).
#pragma unroll
  for (int v = 0; v < 8; ++v) {
    float x = mmin[v];
    x = fminf(x, __shfl_xor(x, 1, 32));
    x = fminf(x, __shfl_xor(x, 2, 32));
    x = fminf(x, __shfl_xor(x, 4, 32));
    x = fminf(x, __shfl_xor(x, 8, 32));
    mmin[v] = x;
  }
  // Row of mmin[v]: v (lower half) or v+8 (upper half). Add ||p||^2 per row, sum.
  const int half = lane >> 4;
  float sum = 0.0f;
#pragma unroll
  for (int v = 0; v < 8; ++v)
    sum += mmin[v] + __shfl(pn_lane, v + 8 * half, 32);
  if ((lane & 15) == 0) atomicAdd(&acc[0], sum);
}

// -------- fem (bottom MSE) kernel -----------------------------------------
__global__ __launch_bounds__(256) void MeshLoss_fem_kernel(
    const float* __restrict__ nm, const float* __restrict__ fem,
    float* __restrict__ acc) {
  float s = 0.0f;
  const int stride = gridDim.x * blockDim.x;
  for (int t = blockIdx.x * blockDim.x + threadIdx.x; t < FEM_COUNT; t += stride) {
    const int k    = t & 31;
    const int tmp  = t >> 5;           // rest*15 + j
    const int j    = tmp % 15;
    const int rest = tmp / 15;         // (b*3 + d)*32 + i
    const size_t off = ((size_t)rest * NM_J + j) * NM_K + k;
    const float d = nm[off] - fem[off];
    s += d * d;
  }
#pragma unroll
  for (int msk = 16; msk >= 1; msk >>= 1) s += __shfl_xor(s, msk, 32);
  if ((threadIdx.x & 31) == 0) atomicAdd(&acc[1], s);
}

// -------- init / finalize --------------------------------------------------
__global__ void MeshLoss_init_kernel(float* acc) { acc[0] = 0.0f; acc[1] = 0.0f; }

__global__ void MeshLoss_fini_kernel(const float* __restrict__ acc,
                                     float* __restrict__ out) {
  out[0] = acc[0] / (float)DIST_COUNT + acc[1] / (float)FEM_COUNT;
}

extern "C" void kernel_launch(void* const* d_in, const int* in_sizes, int n_in,
                              void* d_out, int out_size, void* d_ws, size_t ws_size,
                              hipStream_t stream) {
  const float* network_mesh = (const float*)d_in[0];
  const float* pc           = (const float*)d_in[1];
  const float* fem_mesh     = (const float*)d_in[2];
  float* out = (float*)d_out;
  float* acc = (float*)d_ws;

  MeshLoss_init_kernel<<<1, 1, 0, stream>>>(acc);
  MeshLoss_dist_kernel<<<512, 256, 0, stream>>>(pc, network_mesh, acc);
  MeshLoss_fem_kernel<<<720, 256, 0, stream>>>(network_mesh, fem_mesh, acc);
  MeshLoss_fini_kernel<<<1, 1, 0, stream>>>(acc, out);
}